// Kalman_64476049047613
// MI455X (gfx1250) — compile-verified
//
#include <hip/hip_runtime.h>
#include <math.h>

// CDNA5 / gfx1250 Kalman filter: one wave32 runs the sequential scan,
// all 16x16 fp32 matmuls on the matrix core via V_WMMA_F32_16X16X4_F32.

typedef __attribute__((ext_vector_type(2))) float v2f;
typedef __attribute__((ext_vector_type(8))) float v8f;

#define DN 16
#define LOG2PI 1.8378770664093453f

// C = A(16x16) * B(16x16) accumulated into WMMA C-layout registers.
// A/B read from LDS row-major; TA/TB select transposed indexing.
// f32 A-fragment (16x4): lanes 0-15 hold K={0,1}, lanes 16-31 K={2,3}, M=lane&15.
// f32 B-fragment (4x16): mirrored; N=lane&15.
template <bool TA, bool TB>
__device__ inline v8f mma16(const float* A, const float* B, v8f acc,
                            int half, int lid) {
#pragma unroll
  for (int kk = 0; kk < 4; ++kk) {
    const int k0 = 4 * kk + 2 * half;
    v2f a, b;
    a.x = TA ? A[(k0 + 0) * DN + lid] : A[lid * DN + k0 + 0];
    a.y = TA ? A[(k0 + 1) * DN + lid] : A[lid * DN + k0 + 1];
    b.x = TB ? B[lid * DN + k0 + 0] : B[(k0 + 0) * DN + lid];
    b.y = TB ? B[lid * DN + k0 + 1] : B[(k0 + 1) * DN + lid];
    acc = __builtin_amdgcn_wmma_f32_16x16x4_f32(false, a, false, b,
                                                (short)0, acc, false, false);
  }
  return acc;
}

// Store WMMA C-layout (v8f) to LDS row-major: VGPR r holds M = r + 8*half, N = lid.
__device__ inline void storeC(float* C, v8f acc, int half, int lid) {
#pragma unroll
  for (int rr = 0; rr < 8; ++rr) C[(rr + 8 * half) * DN + lid] = acc[rr];
}

__global__ __launch_bounds__(32) void kalman_wmma_kernel(
    const float* __restrict__ obs,   // T x 16
    const float* __restrict__ Fg,    // 16 x 16
    const float* __restrict__ bg,    // 16
    const float* __restrict__ qg,    // 16
    const float* __restrict__ Hg,    // 16 x 16
    const float* __restrict__ cg,    // 16
    const float* __restrict__ rg,    // 16
    const float* __restrict__ m0g,   // 16
    const float* __restrict__ p0g,   // 16
    float* __restrict__ out_means,   // T x 16
    float* __restrict__ out_covs,    // T x 16 x 16
    float* __restrict__ out_ll,      // 1
    int T) {
  __shared__ alignas(16) float sF[256];    // transition matrix
  __shared__ alignas(16) float sH[256];    // emission matrix
  __shared__ alignas(16) float sP[256];    // filtered covariance (state)
  __shared__ alignas(16) float sFP[256];   // F @ P
  __shared__ alignas(16) float sPP[256];   // predicted covariance
  __shared__ alignas(16) float sHpP[256];  // H @ pP
  __shared__ alignas(16) float sS[256];    // innovation covariance
  __shared__ alignas(16) float sL[256];    // cholesky factor (lower)
  __shared__ alignas(16) float sKT[256];   // K^T = inv(S) @ H @ pP
  __shared__ float sb[16], sc[16], sq2[16], sr2[16], sp02[16];
  __shared__ float sm[16], spm[16], sresid[16], sInvD[16];
  __shared__ float sD2[1];

  const int lane = threadIdx.x;
  const int half = lane >> 4;
  const int lid = lane & 15;

  // ---- load constants into LDS ----
  for (int i = lane; i < 256; i += 32) {
    sF[i] = Fg[i];
    sH[i] = Hg[i];
  }
  if (lane < 16) {
    sb[lane] = bg[lane];
    sc[lane] = cg[lane];
    float q = qg[lane], r = rg[lane], p0 = p0g[lane];
    sq2[lane] = q * q;
    sr2[lane] = r * r;
    sp02[lane] = p0 * p0;
    sm[lane] = m0g[lane];
  }
  __syncthreads();

  float ll = 0.0f;
  const v8f zacc = {0.f, 0.f, 0.f, 0.f, 0.f, 0.f, 0.f, 0.f};

  for (int t = 0; t < T; ++t) {
    v8f pP;
    if (t == 0) {
      // prior is the first "prediction"
      if (lane < 16) spm[lane] = sm[lane];
#pragma unroll
      for (int rr = 0; rr < 8; ++rr)
        pP[rr] = ((rr + 8 * half) == lid) ? sp02[lid] : 0.0f;
    } else {
      // pm = F @ m + b
      if (lane < 16) {
        float s = sb[lane];
#pragma unroll
        for (int j = 0; j < 16; ++j) s += sF[lane * DN + j] * sm[j];
        spm[lane] = s;
      }
      __syncthreads();
      // FP = F @ P
      v8f fp = mma16<false, false>(sF, sP, zacc, half, lid);
      storeC(sFP, fp, half, lid);
      __syncthreads();
      // pP = FP @ F^T + diag(q^2)
      pP = mma16<false, true>(sFP, sF, zacc, half, lid);
#pragma unroll
      for (int rr = 0; rr < 8; ++rr)
        if ((rr + 8 * half) == lid) pP[rr] += sq2[lid];
    }
    storeC(sPP, pP, half, lid);
    __syncthreads();

    // resid = obs - (H @ pm + c)
    if (lane < 16) {
      float s = sc[lane];
#pragma unroll
      for (int j = 0; j < 16; ++j) s += sH[lane * DN + j] * spm[j];
      sresid[lane] = obs[(size_t)t * DN + lane] - s;
    }
    // HpP = H @ pP
    v8f hp = mma16<false, false>(sH, sPP, zacc, half, lid);
    storeC(sHpP, hp, half, lid);
    __syncthreads();
    // S = HpP @ H^T + diag(r^2)
    v8f sv = mma16<false, true>(sHpP, sH, zacc, half, lid);
#pragma unroll
    for (int rr = 0; rr < 8; ++rr)
      if ((rr + 8 * half) == lid) sv[rr] += sr2[lid];
    storeC(sS, sv, half, lid);
    __syncthreads();

    // ---- Cholesky: S = L L^T (column-wise, cooperative) ----
    for (int j = 0; j < 16; ++j) {
      if (lane == j) {
        float s = sS[j * DN + j];
        for (int k = 0; k < j; ++k) {
          float v = sL[j * DN + k];
          s -= v * v;
        }
        float d = sqrtf(s);
        sL[j * DN + j] = d;
        sInvD[j] = 1.0f / d;
      }
      __syncthreads();
      if (lane > j && lane < 16) {
        float s = sS[lane * DN + j];
        for (int k = 0; k < j; ++k) s -= sL[lane * DN + k] * sL[j * DN + k];
        sL[lane * DN + j] = s * sInvD[j];
      }
      __syncthreads();
    }

    // ---- Solve S * K^T = HpP (lane n owns column n); lane 16 solves resid ----
    if (lane < 16) {
      float y[16];
#pragma unroll
      for (int i = 0; i < 16; ++i) {
        float s = sHpP[i * DN + lane];
#pragma unroll
        for (int k = 0; k < i; ++k) s -= sL[i * DN + k] * y[k];
        y[i] = s * sInvD[i];
      }
#pragma unroll
      for (int i = 15; i >= 0; --i) {
        float s = y[i];
#pragma unroll
        for (int k = i + 1; k < 16; ++k) s -= sL[k * DN + i] * y[k];
        y[i] = s * sInvD[i];
      }
#pragma unroll
      for (int i = 0; i < 16; ++i) sKT[i * DN + lane] = y[i];
    } else if (lane == 16) {
      float y[16];
      float d2 = 0.0f;
#pragma unroll
      for (int i = 0; i < 16; ++i) {
        float s = sresid[i];
#pragma unroll
        for (int k = 0; k < i; ++k) s -= sL[i * DN + k] * y[k];
        y[i] = s * sInvD[i];
        d2 += y[i] * y[i];
      }
      sD2[0] = d2;
    }
    __syncthreads();

    // ---- log-likelihood contribution ----
    float lg = (lane < 16) ? logf(sL[lane * DN + lane]) : 0.0f;
#pragma unroll
    for (int off = 16; off > 0; off >>= 1) lg += __shfl_down(lg, off);
    if (lane == 0) ll += -0.5f * 16.0f * LOG2PI - lg - 0.5f * sD2[0];

    // ---- fm = pm + K @ resid  (K[i][j] = KT[j][i]) ----
    if (lane < 16) {
      float s = spm[lane];
#pragma unroll
      for (int j = 0; j < 16; ++j) s += sKT[j * DN + lane] * sresid[j];
      sm[lane] = s;
      out_means[(size_t)t * DN + lane] = s;
    }
    __syncthreads();

    // ---- fP = pP - K @ HpP  (A = K = transpose of KT in LDS) ----
    v8f kh = mma16<true, false>(sKT, sHpP, zacc, half, lid);
    v8f fPv;
#pragma unroll
    for (int rr = 0; rr < 8; ++rr) fPv[rr] = pP[rr] - kh[rr];
    storeC(sP, fPv, half, lid);
    __syncthreads();

    // contiguous 128-bit stores of the covariance tile
    {
      const float4* src = reinterpret_cast<const float4*>(sP);
      float4* dst = reinterpret_cast<float4*>(out_covs + (size_t)t * 256);
      dst[lane * 2 + 0] = src[lane * 2 + 0];
      dst[lane * 2 + 1] = src[lane * 2 + 1];
    }
    __syncthreads();
  }

  if (lane == 0) out_ll[0] = ll;
}

extern "C" void kernel_launch(void* const* d_in, const int* in_sizes, int n_in,
                              void* d_out, int out_size, void* d_ws,
                              size_t ws_size, hipStream_t stream) {
  (void)n_in;
  (void)out_size;
  (void)d_ws;
  (void)ws_size;
  const float* obs = (const float*)d_in[0];
  const float* F = (const float*)d_in[1];
  const float* b = (const float*)d_in[2];
  const float* q = (const float*)d_in[3];
  const float* H = (const float*)d_in[4];
  const float* c = (const float*)d_in[5];
  const float* r = (const float*)d_in[6];
  const float* m0 = (const float*)d_in[7];
  const float* p0 = (const float*)d_in[8];
  const int T = in_sizes[0] / 16;
  float* out = (float*)d_out;
  float* out_means = out;                       // T*16
  float* out_covs = out + (size_t)T * 16;       // T*256
  float* out_ll = out + (size_t)T * 272;        // 1
  kalman_wmma_kernel<<<1, 32, 0, stream>>>(obs, F, b, q, H, c, r, m0, p0,
                                           out_means, out_covs, out_ll, T);
}